// Model_26877905338631
// MI455X (gfx1250) — compile-verified
//
#include <hip/hip_runtime.h>
#include <math.h>

typedef __attribute__((ext_vector_type(2))) float v2f;
typedef __attribute__((ext_vector_type(8))) float v8f;

#define G_   200
#define N0_  512
#define E_   614400
#define K1_  410
#define K2_  328
#define K3_  263
#define FIN_ 6
#define H_   128
#define EPS_ 1e-5f

// ---------------- utility ----------------
__global__ void k_fill_zero(float* p, int n) {
  int i = blockIdx.x * blockDim.x + threadIdx.x;
  if (i < n) p[i] = 0.f;
}

// ---------------- layer 1: F_IN=6 conv (scalar) ----------------
__global__ void k_scatter6(const float* __restrict__ x, const int* __restrict__ src,
                           const int* __restrict__ dst, float* __restrict__ agg) {
  int e = blockIdx.x * blockDim.x + threadIdx.x;
  if (e >= E_) return;
  int s = src[e], d = dst[e];
#pragma unroll
  for (int f = 0; f < FIN_; ++f)
    atomicAdd(&agg[d * FIN_ + f], x[s * FIN_ + f]);
}

__global__ void k_conv1(const float* __restrict__ x, const float* __restrict__ agg,
                        const float* __restrict__ Wrel, const float* __restrict__ Wroot,
                        const float* __restrict__ b, float* __restrict__ out, int nn) {
  int i = blockIdx.x * blockDim.x + threadIdx.x;
  if (i >= nn * H_) return;
  int node = i >> 7, h = i & 127;
  float acc = b[h];
#pragma unroll
  for (int f = 0; f < FIN_; ++f)
    acc += agg[node * FIN_ + f] * Wrel[h * FIN_ + f] + x[node * FIN_ + f] * Wroot[h * FIN_ + f];
  out[i] = fmaxf(acc, 0.f);
}

// ---------------- scoring: tanh((h.w)/||w||), one wave per node ----------------
__global__ void k_score(const float* __restrict__ h, const float* __restrict__ w,
                        float* __restrict__ score, int nn) {
  int tid  = threadIdx.x;
  int lane = tid & 31;
  int node = (blockIdx.x * blockDim.x + tid) >> 5;
  if (node >= nn) return;
  float dot = 0.f, wsq = 0.f;
#pragma unroll
  for (int f = lane; f < H_; f += 32) {
    float wv = w[f];
    dot += h[(size_t)node * H_ + f] * wv;
    wsq += wv * wv;
  }
#pragma unroll
  for (int o = 16; o > 0; o >>= 1) {
    dot += __shfl_xor(dot, o, 32);
    wsq += __shfl_xor(wsq, o, 32);
  }
  if (lane == 0) score[node] = tanhf(dot * rsqrtf(wsq));
}

// ---------------- top-k per graph: bitonic sort of (val desc, idx asc) ----------------
__global__ void k_topk(const float* __restrict__ score, int n, int k,
                       int* __restrict__ perm, int* __restrict__ inv,
                       float* __restrict__ gscore) {
  __shared__ float sv[512];
  __shared__ int   si[512];
  __shared__ int   cnt;
  int g = blockIdx.x;
  int t = threadIdx.x;
  float v = (t < n) ? score[g * n + t] : -INFINITY;
  sv[t] = v;
  si[t] = (t < n) ? t : 0x7FFFFFFF;
  if (t == 0) cnt = 0;
  __syncthreads();
  for (int ks = 2; ks <= 512; ks <<= 1) {
    for (int j = ks >> 1; j > 0; j >>= 1) {
      int ixj = t ^ j;
      if (ixj > t) {
        float v0 = sv[t], v1 = sv[ixj];
        int   i0 = si[t], i1 = si[ixj];
        bool firstBetter = (v0 > v1) || (v0 == v1 && i0 < i1);
        bool up = ((t & ks) == 0);
        if (up ? !firstBetter : firstBetter) {
          sv[t] = v1; si[t] = i1; sv[ixj] = v0; si[ixj] = i0;
        }
      }
      __syncthreads();
    }
  }
  float tv = sv[k - 1];
  int   ti = si[k - 1];
  if (t < n) {
    bool sel = (v > tv) || (v == tv && t <= ti);
    if (sel) {
      int r = atomicAdd(&cnt, 1);        // order within graph is immaterial downstream
      int nid = g * k + r;
      inv[g * n + t] = nid;
      perm[nid]      = g * n + t;
      gscore[nid]    = v;
    } else {
      inv[g * n + t] = -1;
    }
  }
}

__global__ void k_gate(const float* __restrict__ hin, const int* __restrict__ perm,
                       const float* __restrict__ gs, float* __restrict__ hout, int nnew) {
  int i = blockIdx.x * blockDim.x + threadIdx.x;
  if (i >= nnew * H_) return;
  int j = i >> 7, f = i & 127;
  hout[i] = hin[(size_t)perm[j] * H_ + f] * gs[j];
}

__global__ void k_remap(const int* __restrict__ si, const int* __restrict__ di,
                        const int* __restrict__ inv, int* __restrict__ so,
                        int* __restrict__ dout) {
  int e = blockIdx.x * blockDim.x + threadIdx.x;
  if (e >= E_) return;
  int s = si[e];
  int ns = -1, nd = -1;
  if (s >= 0) { ns = inv[s]; nd = inv[di[e]]; }
  if (ns < 0 || nd < 0) { so[e] = -1; dout[e] = 0; }
  else { so[e] = ns; dout[e] = nd; }
}

// ---------------- H=128 message scatter: one wave per edge ----------------
__global__ void k_scatter128(const float* __restrict__ h, const int* __restrict__ src,
                             const int* __restrict__ dst, float* __restrict__ agg) {
  int tid  = blockIdx.x * blockDim.x + threadIdx.x;
  int e    = tid >> 5;
  int lane = tid & 31;
  if (e >= E_) return;
  int s = src[e];
  if (s < 0) return;                 // dead edge (uniform across the wave)
  int d = dst[e];
  const float4* hp = (const float4*)(h + (size_t)s * H_);
  float4 v = hp[lane];
  float* ap = agg + (size_t)d * H_ + lane * 4;
  atomicAdd(ap + 0, v.x); atomicAdd(ap + 1, v.y);
  atomicAdd(ap + 2, v.z); atomicAdd(ap + 3, v.w);
}

// ---------------- dense transform: out = relu(agg@Wrel^T + b + x@Wroot^T), f32 WMMA ----------------
// block = 256 threads = 8 waves; block handles 16 node rows, wave w handles cols [16w,16w+16)
__global__ void k_conv_wmma(const float* __restrict__ agg, const float* __restrict__ xin,
                            const float* __restrict__ Wrel, const float* __restrict__ Wroot,
                            const float* __restrict__ bias, float* __restrict__ out, int nn) {
  int tid  = threadIdx.x;
  int lane = tid & 31;
  int wave = tid >> 5;
  int m0   = blockIdx.x * 16;
  int col0 = wave * 16;
  int half = lane >> 4;   // 0: K pair {0,1}, 1: K pair {2,3}
  int l16  = lane & 15;

  v8f c = {};
  const float* aggp   = agg   + (size_t)(m0 + l16) * H_;
  const float* xp     = xin   + (size_t)(m0 + l16) * H_;
  const float* wrelp  = Wrel  + (size_t)(col0 + l16) * H_;   // B[k][col] = W[col*H + k]
  const float* wrootp = Wroot + (size_t)(col0 + l16) * H_;

  for (int k0 = 0; k0 < H_; k0 += 4) {
    int ka = k0 + (half << 1);
    v2f aA = { aggp[ka],   aggp[ka + 1]   };
    v2f aX = { xp[ka],     xp[ka + 1]     };
    v2f bR = { wrelp[ka],  wrelp[ka + 1]  };
    v2f bT = { wrootp[ka], wrootp[ka + 1] };
    c = __builtin_amdgcn_wmma_f32_16x16x4_f32(false, aA, false, bR, (short)0, c, false, false);
    c = __builtin_amdgcn_wmma_f32_16x16x4_f32(false, aX, false, bT, (short)0, c, false, false);
  }
  float bcol = bias[col0 + l16];
#pragma unroll
  for (int v = 0; v < 8; ++v) {
    int row = m0 + v + (half << 3);
    float val = c[v] + bcol;
    out[(size_t)row * H_ + col0 + l16] = fmaxf(val, 0.f);
  }
}

// ---------------- readout: per-graph max & mean, accumulated into z ----------------
__global__ void k_readout(const float* __restrict__ h, int k, float* __restrict__ z) {
  int i = blockIdx.x * blockDim.x + threadIdx.x;   // G*H threads
  if (i >= G_ * H_) return;
  int g = i >> 7, f = i & 127;
  const float* p = h + ((size_t)g * k) * H_ + f;
  float mx = -INFINITY, sum = 0.f;
  for (int j = 0; j < k; ++j) { float v = p[(size_t)j * H_]; mx = fmaxf(mx, v); sum += v; }
  z[g * 2 * H_ + f]      += mx;
  z[g * 2 * H_ + H_ + f] += sum / (float)k;
}

// ---------------- head ----------------
__global__ void k_fc(const float* __restrict__ in, const float* __restrict__ W,
                     const float* __restrict__ b, float* __restrict__ out,
                     int rows, int cin, int cout) {
  int i = blockIdx.x * blockDim.x + threadIdx.x;
  if (i >= rows * cout) return;
  int r = i / cout, o = i % cout;
  const float* ip = in + (size_t)r * cin;
  const float* wp = W  + (size_t)o * cin;
  float acc = b[o];
  for (int c = 0; c < cin; ++c) acc += ip[c] * wp[c];
  out[i] = acc;
}

__global__ void k_bn_relu(float* __restrict__ y, const float* __restrict__ gamma,
                          const float* __restrict__ beta, int cols) {
  __shared__ float rs[256], rss[256];
  int col = blockIdx.x;
  int t   = threadIdx.x;
  float s = 0.f, ss = 0.f;
  for (int r = t; r < G_; r += 256) { float v = y[r * cols + col]; s += v; ss += v * v; }
  rs[t] = s; rss[t] = ss;
  __syncthreads();
  for (int o = 128; o > 0; o >>= 1) {
    if (t < o) { rs[t] += rs[t + o]; rss[t] += rss[t + o]; }
    __syncthreads();
  }
  float mean = rs[0] / (float)G_;
  float var  = rss[0] / (float)G_ - mean * mean;
  float istd = rsqrtf(var + EPS_);
  float gm = gamma[col], bt = beta[col];
  for (int r = t; r < G_; r += 256) {
    float v = y[r * cols + col];
    y[r * cols + col] = fmaxf((v - mean) * istd * gm + bt, 0.f);
  }
}

__global__ void k_head(const float* __restrict__ y2, const float* __restrict__ W,
                       const float* __restrict__ b, float* __restrict__ out) {
  int g = blockIdx.x * blockDim.x + threadIdx.x;
  if (g >= G_) return;
  float l0 = b[0], l1 = b[1];
  const float* p = y2 + g * 64;
  for (int c = 0; c < 64; ++c) { l0 += p[c] * W[c]; l1 += p[c] * W[64 + c]; }
  float m  = fmaxf(l0, l1);
  float e0 = __expf(l0 - m), e1 = __expf(l1 - m);
  float iv = 1.f / (e0 + e1);
  out[g * 2 + 0] = e0 * iv;
  out[g * 2 + 1] = e1 * iv;
}

// ---------------- launcher ----------------
extern "C" void kernel_launch(void* const* d_in, const int* in_sizes, int n_in,
                              void* d_out, int out_size, void* d_ws, size_t ws_size,
                              hipStream_t stream) {
  (void)in_sizes; (void)n_in; (void)out_size; (void)ws_size;
  const float* x      = (const float*)d_in[0];
  const int*   ei     = (const int*)d_in[1];
  const float* Wrel1  = (const float*)d_in[2];
  const float* Wroot1 = (const float*)d_in[3];
  const float* b1     = (const float*)d_in[4];
  const float* Wrel2  = (const float*)d_in[5];
  const float* Wroot2 = (const float*)d_in[6];
  const float* b2     = (const float*)d_in[7];
  const float* Wrel3  = (const float*)d_in[8];
  const float* Wroot3 = (const float*)d_in[9];
  const float* b3     = (const float*)d_in[10];
  const float* pw1    = (const float*)d_in[11];
  const float* pw2    = (const float*)d_in[12];
  const float* pw3    = (const float*)d_in[13];
  const float* lin1_w = (const float*)d_in[14];
  const float* lin1_b = (const float*)d_in[15];
  const float* lin2_w = (const float*)d_in[16];
  const float* lin2_b = (const float*)d_in[17];
  const float* lin3_w = (const float*)d_in[18];
  const float* lin3_b = (const float*)d_in[19];
  const float* bn1_g  = (const float*)d_in[20];
  const float* bn1_b  = (const float*)d_in[21];
  const float* bn2_g  = (const float*)d_in[22];
  const float* bn2_b  = (const float*)d_in[23];
  float* out = (float*)d_out;

  char* ws = (char*)d_ws;
  size_t off = 0;
  auto alloc = [&](size_t bytes) -> void* {
    void* p = ws + off;
    off = (off + bytes + 255) & ~(size_t)255;
    return p;
  };
  float* bufA   = (float*)alloc(sizeof(float) * (size_t)102400 * 128); // h0 -> conv2out -> h3
  float* bufB   = (float*)alloc(sizeof(float) * (size_t)102400 * 128); // agg scratch
  float* bufC   = (float*)alloc(sizeof(float) * (size_t)82000 * 128);  // h1 -> conv3out
  float* bufD   = (float*)alloc(sizeof(float) * (size_t)65600 * 128);  // h2
  float* score  = (float*)alloc(sizeof(float) * 102400);
  float* gscore = (float*)alloc(sizeof(float) * 82000);
  int*   perm   = (int*)alloc(sizeof(int) * 82000);
  int*   inv    = (int*)alloc(sizeof(int) * 102400);
  int*   srcA   = (int*)alloc(sizeof(int) * E_);
  int*   dstA   = (int*)alloc(sizeof(int) * E_);
  int*   srcB   = (int*)alloc(sizeof(int) * E_);
  int*   dstB   = (int*)alloc(sizeof(int) * E_);
  float* z      = (float*)alloc(sizeof(float) * G_ * 256);
  float* y1     = (float*)alloc(sizeof(float) * G_ * 128);
  float* y2     = (float*)alloc(sizeof(float) * G_ * 64);

  const int* esrc = ei;
  const int* edst = ei + E_;

  // ---- Level 1 (F_IN=6 -> H) ----
  k_fill_zero<<<(102400 * FIN_ + 255) / 256, 256, 0, stream>>>(bufB, 102400 * FIN_);
  k_scatter6<<<(E_ + 255) / 256, 256, 0, stream>>>(x, esrc, edst, bufB);
  k_conv1<<<(102400 * H_ + 255) / 256, 256, 0, stream>>>(x, bufB, Wrel1, Wroot1, b1, bufA, 102400);
  k_score<<<(102400 * 32 + 255) / 256, 256, 0, stream>>>(bufA, pw1, score, 102400);
  k_topk<<<G_, 512, 0, stream>>>(score, N0_, K1_, perm, inv, gscore);
  k_gate<<<(G_ * K1_ * H_ + 255) / 256, 256, 0, stream>>>(bufA, perm, gscore, bufC, G_ * K1_);
  k_remap<<<(E_ + 255) / 256, 256, 0, stream>>>(esrc, edst, inv, srcA, dstA);
  k_fill_zero<<<(G_ * 256 + 255) / 256, 256, 0, stream>>>(z, G_ * 256);
  k_readout<<<(G_ * H_ + 255) / 256, 256, 0, stream>>>(bufC, K1_, z);

  // ---- Level 2 (H -> H, WMMA) ----
  k_fill_zero<<<(G_ * K1_ * H_ + 255) / 256, 256, 0, stream>>>(bufB, G_ * K1_ * H_);
  k_scatter128<<<(E_ * 32 + 255) / 256, 256, 0, stream>>>(bufC, srcA, dstA, bufB);
  k_conv_wmma<<<(G_ * K1_) / 16, 256, 0, stream>>>(bufB, bufC, Wrel2, Wroot2, b2, bufA, G_ * K1_);
  k_score<<<(G_ * K1_ * 32 + 255) / 256, 256, 0, stream>>>(bufA, pw2, score, G_ * K1_);
  k_topk<<<G_, 512, 0, stream>>>(score, K1_, K2_, perm, inv, gscore);
  k_gate<<<(G_ * K2_ * H_ + 255) / 256, 256, 0, stream>>>(bufA, perm, gscore, bufD, G_ * K2_);
  k_remap<<<(E_ + 255) / 256, 256, 0, stream>>>(srcA, dstA, inv, srcB, dstB);
  k_readout<<<(G_ * H_ + 255) / 256, 256, 0, stream>>>(bufD, K2_, z);

  // ---- Level 3 (H -> H, WMMA) ----
  k_fill_zero<<<(G_ * K2_ * H_ + 255) / 256, 256, 0, stream>>>(bufB, G_ * K2_ * H_);
  k_scatter128<<<(E_ * 32 + 255) / 256, 256, 0, stream>>>(bufD, srcB, dstB, bufB);
  k_conv_wmma<<<(G_ * K2_) / 16, 256, 0, stream>>>(bufB, bufD, Wrel3, Wroot3, b3, bufC, G_ * K2_);
  k_score<<<(G_ * K2_ * 32 + 255) / 256, 256, 0, stream>>>(bufC, pw3, score, G_ * K2_);
  k_topk<<<G_, 512, 0, stream>>>(score, K2_, K3_, perm, inv, gscore);
  k_gate<<<(G_ * K3_ * H_ + 255) / 256, 256, 0, stream>>>(bufC, perm, gscore, bufA, G_ * K3_);
  k_readout<<<(G_ * H_ + 255) / 256, 256, 0, stream>>>(bufA, K3_, z);

  // ---- Head ----
  k_fc<<<(G_ * 128 + 255) / 256, 256, 0, stream>>>(z, lin1_w, lin1_b, y1, G_, 256, 128);
  k_bn_relu<<<128, 256, 0, stream>>>(y1, bn1_g, bn1_b, 128);
  k_fc<<<(G_ * 64 + 255) / 256, 256, 0, stream>>>(y1, lin2_w, lin2_b, y2, G_, 128, 64);
  k_bn_relu<<<64, 256, 0, stream>>>(y2, bn2_g, bn2_b, 64);
  k_head<<<(G_ + 255) / 256, 256, 0, stream>>>(y2, lin3_w, lin3_b, out);
}